// CausalDepthwiseConv1d_21449066676463
// MI455X (gfx1250) — compile-verified
//
#include <hip/hip_runtime.h>
#include <stdint.h>

typedef __attribute__((ext_vector_type(4))) float v4f;

namespace {
constexpr int KK      = 4;      // conv taps
constexpr int Bsz     = 4;
constexpr int Tsz     = 4096;
constexpr int Csz     = 2048;
constexpr int THREADS = 256;    // 8 waves (wave32)
constexpr int CT      = THREADS * 4;   // 1024 channels per workgroup
constexpr int LT      = 64;            // timesteps per workgroup
constexpr int DEPTH   = 4;             // async pipeline depth (ring of LDS buffers)
}

// gfx1250 async global->LDS copy, GVS mode: uniform 64-bit SGPR base + constant
// per-lane 32-bit VGPR offset -> per-copy address math is SALU-only.
#define ASYNC_COPY_B128(ldsoff, voff, sbase)                              \
    asm volatile("global_load_async_to_lds_b128 %0, %1, %2"               \
                 :: "v"(ldsoff), "v"(voff), "s"(sbase) : "memory")

#define WAIT_ASYNCCNT(n) asm volatile("s_wait_asynccnt " #n ::: "memory")
#define WAIT_DSCNT0()    asm volatile("s_wait_dscnt 0x0" ::: "memory")

__global__ __launch_bounds__(THREADS)
void causal_dwconv_fused(const float* __restrict__ x,
                         const float* __restrict__ W,
                         float* __restrict__ y,
                         float* __restrict__ cache)
{
    __shared__ float buf[DEPTH][CT];   // 4 x 4KB ring buffer

    const int tid  = threadIdx.x;
    const int b    = blockIdx.z;
    const int cblk = blockIdx.y * CT;
    const int c    = cblk + tid * 4;           // this thread's 4 channels
    const int t0   = blockIdx.x * LT;

    // per-channel taps: W[c+j, 0..3] are 16B contiguous
    const v4f w0 = *(const v4f*)(W + (size_t)(c + 0) * KK);
    const v4f w1 = *(const v4f*)(W + (size_t)(c + 1) * KK);
    const v4f w2 = *(const v4f*)(W + (size_t)(c + 2) * KK);
    const v4f w3 = *(const v4f*)(W + (size_t)(c + 3) * KK);

    const float* xrow = x + ((size_t)b * Tsz) * Csz + cblk;  // uniform row base (saddr)
    const float* xb   = x + ((size_t)b * Tsz) * Csz + c;     // per-lane (halo loads)
    float*       yb   = y + ((size_t)b * Tsz) * Csz + c;

    // register sliding window: x[t-1], x[t-2], x[t-3] (zeros left of t=0)
    v4f xm1 = {0.f, 0.f, 0.f, 0.f};
    v4f xm2 = {0.f, 0.f, 0.f, 0.f};
    v4f xm3 = {0.f, 0.f, 0.f, 0.f};
    if (t0 >= 1) xm1 = *(const v4f*)(xb + (size_t)(t0 - 1) * Csz);
    if (t0 >= 2) xm2 = *(const v4f*)(xb + (size_t)(t0 - 2) * Csz);
    if (t0 >= 3) xm3 = *(const v4f*)(xb + (size_t)(t0 - 3) * Csz);

    // LDS byte offset of this lane's 16B slot in ring buffer 0
    // (flat->LDS mapping truncates to addr[31:0] = LDS offset, ISA 10.2)
    const uint32_t lds0 = (uint32_t)(uintptr_t)(&buf[0][0]) + (uint32_t)(tid * 16);
    const uint32_t voff = (uint32_t)(tid * 16);  // constant per-lane GVS offset

    // prime the pipeline: DEPTH rows in flight
    #pragma unroll
    for (int d = 0; d < DEPTH; ++d) {
        ASYNC_COPY_B128(lds0 + (uint32_t)(d * CT * 4), voff,
                        (unsigned long long)(uintptr_t)(xrow + (size_t)(t0 + d) * Csz));
    }

    // one conv step: consume current row, NT-store y, advance window
    auto stepc = [&](const v4f cur, const int t) {
        v4f o;
        o.x = w0.x * xm3.x + w0.y * xm2.x + w0.z * xm1.x + w0.w * cur.x;
        o.y = w1.x * xm3.y + w1.y * xm2.y + w1.z * xm1.y + w1.w * cur.y;
        o.z = w2.x * xm3.z + w2.y * xm2.z + w2.z * xm1.z + w2.w * cur.z;
        o.w = w3.x * xm3.w + w3.y * xm2.w + w3.z * xm1.w + w3.w * cur.w;
        __builtin_nontemporal_store(o, (v4f*)(yb + (size_t)t * Csz));
        xm3 = xm2; xm2 = xm1; xm1 = cur;
    };

    // steady state: keep DEPTH copies in flight; (i & 3) is constant per
    // unrolled-by-4 body so LDS addresses fold to immediates
    #pragma unroll 4
    for (int i = 0; i < LT - DEPTH; ++i) {
        WAIT_ASYNCCNT(3);                     // copy for row t0+i complete
        const v4f cur = *(const v4f*)(&buf[i & 3][tid * 4]);
        WAIT_DSCNT0();                        // ds read retired before slot re-arm (WAR)
        ASYNC_COPY_B128(lds0 + (uint32_t)((i & 3) * CT * 4), voff,
                        (unsigned long long)(uintptr_t)(xrow + (size_t)(t0 + i + DEPTH) * Csz));
        stepc(cur, t0 + i);
    }

    // tail: drain the ring (remaining outstanding = 3,2,1,0)
    {
        const int jb = LT - DEPTH;
        WAIT_ASYNCCNT(3);
        stepc(*(const v4f*)(&buf[(jb + 0) & 3][tid * 4]), t0 + jb + 0);
        WAIT_ASYNCCNT(2);
        stepc(*(const v4f*)(&buf[(jb + 1) & 3][tid * 4]), t0 + jb + 1);
        WAIT_ASYNCCNT(1);
        stepc(*(const v4f*)(&buf[(jb + 2) & 3][tid * 4]), t0 + jb + 2);
        WAIT_ASYNCCNT(0);
        stepc(*(const v4f*)(&buf[(jb + 3) & 3][tid * 4]), t0 + jb + 3);
    }

    // new_cache[b, c, k] = x[b, T-3+k, c] straight from the register window
    if (blockIdx.x == gridDim.x - 1) {
        #pragma unroll
        for (int j = 0; j < 4; ++j) {
            float* cp = cache + ((size_t)b * Csz + (size_t)(c + j)) * (KK - 1);
            cp[0] = xm3[j];
            cp[1] = xm2[j];
            cp[2] = xm1[j];
        }
    }
}

extern "C" void kernel_launch(void* const* d_in, const int* in_sizes, int n_in,
                              void* d_out, int out_size, void* d_ws, size_t ws_size,
                              hipStream_t stream) {
    const float* x = (const float*)d_in[0];   // [B, T, C] fp32
    const float* W = (const float*)d_in[1];   // [C, K]   fp32
    float* y     = (float*)d_out;                          // [B, T, C]
    float* cache = y + (size_t)Bsz * Tsz * Csz;            // [B, C, K-1]

    dim3 grid(Tsz / LT, Csz / CT, Bsz);   // (64, 2, 4) = 512 workgroups
    causal_dwconv_fused<<<grid, dim3(THREADS), 0, stream>>>(x, W, y, cache);

    (void)in_sizes; (void)n_in; (void)out_size; (void)d_ws; (void)ws_size;
}